// LightNet_82617990906648
// MI455X (gfx1250) — compile-verified
//
#include <hip/hip_runtime.h>
#include <math.h>

// LightNet: low-pass (rank-6 Fourier projection, CUTOFF=3) + 1x1-conv MLP + residual + clip.
// B=32, C=3, H=W=512.  Kept freqs per axis: {0,1,2,-3,-2,-1} (6 complex coeffs).
//
// Pipeline (all f32, memory-bound ~200MB total traffic, HBM floor ~9us @ 23.3 TB/s):
//   k_trig     : table[k][12] = {cos, sin}(2*pi*s_j*k/512) for the 6 signed freqs
//   k_rowfft   : P[row][16] (12 used) = row-wise projection of x  -- V_WMMA_F32_16X16X4_F32 GEMM,
//                x staged to LDS via GLOBAL_LOAD_ASYNC_TO_LDS_B128, double-buffered
//   k_colfft   : F[img][6][6] complex  (36-coeff reduction over h)
//   k_rowsynth : T[img][h][6] complex = (1/HW) * F * e^{+i w u h}
//   k_out      : lp = |sum_v T[h][v] e^{+i w v w}| ; fused 3->8 relu ->3 MLP + lp, clip [0,1]

#define HH 512
#define WW 512
#define BB 32
#define CC 3
#define NCOL 12               // 6 complex freqs = 12 reals
#define NROWS (BB*CC*HH)      // 49152 rows for the stage-1 GEMM
#define NIMG  (BB*CC)         // 96 images

#define CHUNK 64              // K-chunk staged in LDS per wave
#define STRD  68              // LDS row stride (floats): banks 4m+k2 -> conflict-free b64 reads
#define STGF  (16*STRD)       // floats per staging buffer (1088)

typedef __attribute__((ext_vector_type(2))) float v2f;
typedef __attribute__((ext_vector_type(4))) float v4f;
typedef __attribute__((ext_vector_type(8))) float v8f;
typedef __attribute__((ext_vector_type(4))) int   v4i;

#if __has_builtin(__builtin_amdgcn_global_load_async_to_lds_b128) && \
    __has_builtin(__builtin_amdgcn_s_wait_asynccnt)
#define USE_ASYNC 1
typedef __attribute__((address_space(1))) v4i v4i_g;   // global (AS1) int4
typedef __attribute__((address_space(3))) v4i v4i_l;   // LDS (AS3) int4
#else
#define USE_ASYNC 0
#endif

__global__ void k_trig(float* __restrict__ trig) {
  int k = blockIdx.x * blockDim.x + threadIdx.x;
  if (k >= HH) return;
  const float w0 = 6.28318530717958647692f / (float)HH;
  #pragma unroll
  for (int j = 0; j < 6; ++j) {
    int s = (j < 3) ? j : j - 6;          // 0,1,2,-3,-2,-1
    float th = w0 * (float)(s * k);
    float sn, cs;
    sincosf(th, &sn, &cs);
    trig[k * NCOL + 2 * j]     = cs;
    trig[k * NCOL + 2 * j + 1] = sn;
  }
}

// Issue the coalesced copy of one 16 x CHUNK x f32 chunk into this wave's staging buffer.
// 8 instructions x 32 lanes x 16B: lane -> (row = 2*it + lane/16, col4 = (lane&15)*4).
__device__ __forceinline__ void copy_chunk(const float* __restrict__ x0, float* stg,
                                           int kc, int lane) {
  int r2 = lane >> 4;
  int c4 = (lane & 15) * 4;
  #pragma unroll
  for (int it = 0; it < 8; ++it) {
    int row = it * 2 + r2;
    const float* g = x0 + (size_t)row * WW + kc + c4;
    float* l = stg + row * STRD + c4;
#if USE_ASYNC
    __builtin_amdgcn_global_load_async_to_lds_b128(
        (v4i_g*)(v4i*)g, (v4i_l*)(v4i*)l, 0, 0);
#else
    *(v4f*)l = *(const v4f*)g;
#endif
  }
}

// Stage 1: P[row][n] = sum_k x[row][k] * B[k][n],
//   B[k][2j] = cos(w s_j k), B[k][2j+1] = -sin(w s_j k)  (forward DFT: e^{-i th})
// 4 waves/block, one 16-row tile per wave. K=512 in 8 chunks of 64, WMMA f32 16x16x4.
__global__ __launch_bounds__(128) void k_rowfft(const float* __restrict__ x,
                                                const float* __restrict__ trig,
                                                float* __restrict__ P) {
  __shared__ float smem[HH * NCOL + 4 * 2 * STGF];   // 24KB basis + 34KB staging (<64KB)
  int tid = threadIdx.x;
  for (int i = tid; i < HH * NCOL; i += 128) {       // basis: trig with Im negated
    float v = trig[i];
    if (i & 1) v = -v;
    smem[i] = v;
  }
  __syncthreads();
  const float* sB = smem;

  int wave = tid >> 5;
  int lane = tid & 31;
  int tile = blockIdx.x * 4 + wave;                  // 3072 tiles total
  int row0 = tile * 16;
  int m  = lane & 15;                                // A fragment row
  int hl = lane >> 4;                                // lane half -> K sub-pair
  int n  = lane & 15;                                // B/D fragment column
  int nn = (n < NCOL) ? n : 0;                       // clamp: padded N cols read col 0, zeroed
  const float* x0 = x + (size_t)row0 * WW;
  float* stg = smem + HH * NCOL + wave * (2 * STGF);

  copy_chunk(x0, stg, 0, lane);

  v8f acc = {0.f, 0.f, 0.f, 0.f, 0.f, 0.f, 0.f, 0.f};
  for (int c = 0; c < 8; ++c) {
    int kc = c * CHUNK;
    if (c < 7) copy_chunk(x0, stg + ((c + 1) & 1) * STGF, kc + CHUNK, lane);
#if USE_ASYNC
    if (c < 7) __builtin_amdgcn_s_wait_asynccnt(8);  // in-order: chunk c retired
    else       __builtin_amdgcn_s_wait_asynccnt(0);
#endif
    const float* sA = stg + (c & 1) * STGF;
    #pragma unroll
    for (int k = 0; k < CHUNK; k += 4) {
      int k2 = k + 2 * hl;                           // A/B layout: VGPR j holds K=k+j+2*half
      v2f a = *(const v2f*)&sA[m * STRD + k2];
      int kg = kc + k2;
      float bx = sB[kg * NCOL + nn];
      float by = sB[(kg + 1) * NCOL + nn];
      if (n >= NCOL) { bx = 0.f; by = 0.f; }
      v2f b; b.x = bx; b.y = by;
      acc = __builtin_amdgcn_wmma_f32_16x16x4_f32(
          /*neg_a=*/false, a, /*neg_b=*/false, b,
          /*c_mod=*/(short)0, acc, /*reuse_a=*/false, /*reuse_b=*/false);
    }
  }
  // D layout: VGPR r -> M = r + 8*half, N = lane&15
  #pragma unroll
  for (int r = 0; r < 8; ++r) {
    int mr = r + 8 * hl;
    P[(size_t)(row0 + mr) * 16 + n] = acc[r];
  }
}

// Stage 2: F[u,v] = sum_h P[h,v] * e^{-i w s_u h}   (complex * complex)
__global__ void k_colfft(const float* __restrict__ P, const float* __restrict__ trig,
                         float* __restrict__ F) {
  int img = blockIdx.x;                   // 96
  int t = threadIdx.x;                    // 64 threads, 36 active
  if (t >= 36) return;
  int up = t / 6, vp = t % 6;
  const float* Pi0 = P + (size_t)img * HH * 16;
  float fr = 0.f, fi = 0.f;
  for (int h = 0; h < HH; ++h) {
    float pr = Pi0[h * 16 + 2 * vp];
    float pi = Pi0[h * 16 + 2 * vp + 1];
    float c  = trig[h * NCOL + 2 * up];
    float s  = trig[h * NCOL + 2 * up + 1];
    fr += pr * c + pi * s;                // (pr + i pi)(c - i s)
    fi += pi * c - pr * s;
  }
  F[(size_t)img * 72 + 2 * t]     = fr;
  F[(size_t)img * 72 + 2 * t + 1] = fi;
}

// Stage 3: T[h,v] = (1/HW) sum_u F[u,v] * e^{+i w s_u h}
__global__ void k_rowsynth(const float* __restrict__ F, const float* __restrict__ trig,
                           float* __restrict__ T) {
  int gid = blockIdx.x * blockDim.x + threadIdx.x;   // NIMG*HH
  if (gid >= NIMG * HH) return;
  int img = gid >> 9, h = gid & (HH - 1);
  const float scale = 1.0f / ((float)HH * (float)WW);
  const float* Fi0 = F + (size_t)img * 72;
  #pragma unroll
  for (int vp = 0; vp < 6; ++vp) {
    float tr = 0.f, ti = 0.f;
    #pragma unroll
    for (int up = 0; up < 6; ++up) {
      float fr = Fi0[(up * 6 + vp) * 2];
      float fi = Fi0[(up * 6 + vp) * 2 + 1];
      float c  = trig[h * NCOL + 2 * up];
      float s  = trig[h * NCOL + 2 * up + 1];
      tr += fr * c - fi * s;              // (fr + i fi)(c + i s)
      ti += fr * s + fi * c;
    }
    T[(size_t)gid * NCOL + 2 * vp]     = tr * scale;
    T[(size_t)gid * NCOL + 2 * vp + 1] = ti * scale;
  }
}

// Stage 4: lp = |sum_v T[h,v] e^{+i w s_v w}| ; fused MLP 3->8(relu)->3 + lp, clip [0,1]
__global__ __launch_bounds__(256) void k_out(const float* __restrict__ T,
                                             const float* __restrict__ trig,
                                             const float* __restrict__ w1,
                                             const float* __restrict__ b1,
                                             const float* __restrict__ w2,
                                             const float* __restrict__ b2,
                                             float* __restrict__ out) {
  int bh = blockIdx.x;                    // b*512 + h
  int b = bh >> 9, h = bh & (HH - 1);

  float Tr[CC][6], Ti[CC][6];
  #pragma unroll
  for (int c = 0; c < CC; ++c) {
    const float* tp = T + (size_t)((b * CC + c) * HH + h) * NCOL;
    #pragma unroll
    for (int v = 0; v < 6; ++v) { Tr[c][v] = tp[2 * v]; Ti[c][v] = tp[2 * v + 1]; }
  }

  for (int w = threadIdx.x; w < WW; w += blockDim.x) {
    float cs[6], sn[6];
    #pragma unroll
    for (int v = 0; v < 6; ++v) {
      cs[v] = trig[w * NCOL + 2 * v];
      sn[v] = trig[w * NCOL + 2 * v + 1];
    }
    float lp[CC];
    #pragma unroll
    for (int c = 0; c < CC; ++c) {
      float zr = 0.f, zi = 0.f;
      #pragma unroll
      for (int v = 0; v < 6; ++v) {
        zr += Tr[c][v] * cs[v] - Ti[c][v] * sn[v];
        zi += Tr[c][v] * sn[v] + Ti[c][v] * cs[v];
      }
      lp[c] = sqrtf(zr * zr + zi * zi);
    }
    float hf[8];
    #pragma unroll
    for (int f = 0; f < 8; ++f) {
      float a = b1[f];
      #pragma unroll
      for (int c = 0; c < CC; ++c) a += w1[f * CC + c] * lp[c];
      hf[f] = a > 0.f ? a : 0.f;
    }
    #pragma unroll
    for (int c = 0; c < CC; ++c) {
      float y = b2[c] + lp[c];
      #pragma unroll
      for (int f = 0; f < 8; ++f) y += w2[c * 8 + f] * hf[f];
      y = y < 0.f ? 0.f : (y > 1.f ? 1.f : y);
      out[((size_t)(b * CC + c) * HH + h) * WW + w] = y;
    }
  }
}

extern "C" void kernel_launch(void* const* d_in, const int* in_sizes, int n_in,
                              void* d_out, int out_size, void* d_ws, size_t ws_size,
                              hipStream_t stream) {
  const float* x  = (const float*)d_in[0];
  const float* w1 = (const float*)d_in[1];
  const float* b1 = (const float*)d_in[2];
  const float* w2 = (const float*)d_in[3];
  const float* b2 = (const float*)d_in[4];
  float* out = (float*)d_out;

  float* ws   = (float*)d_ws;
  float* trig = ws;                       // 512*12 floats (padded region: 8192)
  float* P    = ws + 8192;                // 49152*16 floats
  float* F    = P + (size_t)NROWS * 16;   // 96*72 floats
  float* T    = F + (size_t)NIMG * 72;    // 96*512*12 floats   (total ~5.6 MB)

  k_trig    <<<2,          256, 0, stream>>>(trig);
  k_rowfft  <<<NROWS/16/4, 128, 0, stream>>>(x, trig, P);   // 768 blocks, 4 waves each
  k_colfft  <<<NIMG,       64,  0, stream>>>(P, trig, F);
  k_rowsynth<<<(NIMG*HH + 255)/256, 256, 0, stream>>>(F, trig, T);
  k_out     <<<BB*HH,      256, 0, stream>>>(T, trig, w1, b1, w2, b2, out);
}